// SynonymModule_63239098466878
// MI455X (gfx1250) — compile-verified
//
#include <hip/hip_runtime.h>
#include <hip/hip_bf16.h>
#include <stdint.h>

typedef __attribute__((ext_vector_type(2))) float v2f;
typedef __attribute__((ext_vector_type(4))) float v4f;
typedef __attribute__((ext_vector_type(8))) float v8f;
typedef __attribute__((ext_vector_type(4))) int   v4i;

#define AS1 __attribute__((address_space(1)))
#define AS3 __attribute__((address_space(3)))

static constexpr int T    = 1024;   // tokens
static constexpr int EFF  = 758;    // EFFECT_DIM
static constexpr int ADDD = 10;     // ADD_DIM
static constexpr int KDIM = 768;    // EMBED_DIM
static constexpr int NDIM = 95000;  // MAX_ID

static constexpr int BM   = 256;     // rows per block (16 m-tiles)
static constexpr int BN   = 64;      // cols per block (4 n-tiles)
static constexpr int BK   = 64;      // K-chunk
static constexpr int LDA  = BK + 4;  // 68: conflict-free b64 A-frag reads
static constexpr int LDBT = BK + 4;  // 68: B stored transposed [n][k], same pattern

// ---------- async global -> LDS copies (CDNA5), with safe fallbacks -------
__device__ __forceinline__ void async_cp16(float* l, const float* g) {
#if __has_builtin(__builtin_amdgcn_global_load_async_to_lds_b128)
    __builtin_amdgcn_global_load_async_to_lds_b128(
        (AS1 v4i*)const_cast<float*>(g), (AS3 v4i*)l, 0, 0);
#else
    *(v4f*)l = *(const v4f*)g;
#endif
}

__device__ __forceinline__ void async_cp4(float* l, const float* g) {
#if __has_builtin(__builtin_amdgcn_global_load_async_to_lds_b32)
    __builtin_amdgcn_global_load_async_to_lds_b32(
        (AS1 int*)const_cast<float*>(g), (AS3 int*)l, 0, 0);
#else
    *l = *g;
#endif
}

__device__ __forceinline__ void wait_async() {
#if __has_builtin(__builtin_amdgcn_s_wait_asynccnt)
    __builtin_amdgcn_s_wait_asynccnt(0);
#else
    asm volatile("s_wait_asynccnt 0" ::: "memory");
#endif
}

// ---------------- Phase 1: build X[1024 x 768] into workspace -------------
__global__ __launch_bounds__(256) void build_x_kernel(
    const int*   __restrict__ ids,
    const float* __restrict__ emb_table,
    const float* __restrict__ syn_table,
    const float* __restrict__ W_syn,
    const float* __restrict__ b_syn,
    float*       __restrict__ X)
{
    __shared__ float syn_s[ADDD];
    const int t  = blockIdx.x;
    const int id = ids[t];
    if (threadIdx.x < ADDD) syn_s[threadIdx.x] = syn_table[(long)id * ADDD + threadIdx.x];
    __syncthreads();

    for (int c = threadIdx.x; c < KDIM; c += 256) {
        float v;
        if (c < EFF) {
            v = emb_table[(long)id * EFF + c] + b_syn[c];
#pragma unroll
            for (int k = 0; k < ADDD; ++k)
                v = fmaf(syn_s[k], W_syn[c * ADDD + k], v);
        } else {
            v = 0.1f;
        }
        X[t * KDIM + c] = v;
    }
}

// ---------------- Phase 2: X[1024x768] @ W[768x95000] -> out --------------
__global__ __launch_bounds__(256) void gemm_kernel(
    const float* __restrict__ X,
    const float* __restrict__ W,
    float*       __restrict__ out)
{
    __shared__ float As [BM * LDA];   // 69632 B, row-major [m][k]
    __shared__ float BsT[BN * LDBT];  // 17408 B, transposed [n][k]

    const int tid   = threadIdx.x;
    const int wave  = tid >> 5;
    const int lane  = tid & 31;
    const int lmod  = lane & 15;
    const int lhalf = lane >> 4;

    const int mbase  = blockIdx.x * BM;      // always full (1024 rows total)
    const int nbase  = blockIdx.y * BN;
    const bool full_n = (nbase + BN) <= NDIM;

    v8f acc[2][4];
    const v8f zero = {0.f, 0.f, 0.f, 0.f, 0.f, 0.f, 0.f, 0.f};
#pragma unroll
    for (int mt = 0; mt < 2; ++mt)
#pragma unroll
        for (int nt = 0; nt < 4; ++nt)
            acc[mt][nt] = zero;

    for (int kc = 0; kc < KDIM; kc += BK) {
        __syncthreads();   // previous chunk's compute done before overwrite

        // Stage A: rows mbase..mbase+255, cols kc..kc+63. Thread = one row.
        {
            const float* g = X + (long)(mbase + tid) * KDIM + kc;
            float*       l = As + tid * LDA;
#pragma unroll
            for (int j = 0; j < 16; ++j)
                async_cp16(l + j * 4, g + j * 4);
        }
        // Stage B transposed: BsT[n*LDBT + k] = W[kc+k][nbase+n].
        // Consecutive lanes read consecutive n (coalesced); async unit
        // scatters per-lane into the transposed LDS layout.
        if (full_n) {
#pragma unroll
            for (int i = 0; i < 16; ++i) {
                const int idx = tid + i * 256;   // 0..4095
                const int n   = idx & 63;
                const int k   = idx >> 6;
                async_cp4(BsT + n * LDBT + k,
                          W + (long)(kc + k) * NDIM + nbase + n);
            }
        } else {  // last N strip: predicated scalar staging
#pragma unroll
            for (int i = 0; i < 16; ++i) {
                const int idx = tid + i * 256;
                const int n   = idx & 63;
                const int k   = idx >> 6;
                const int gn  = nbase + n;
                BsT[n * LDBT + k] =
                    (gn < NDIM) ? W[(long)(kc + k) * NDIM + gn] : 0.f;
            }
        }
        wait_async();
        __syncthreads();

        // 16 K-steps of V_WMMA_F32_16X16X4_F32; all frags are b64 LDS loads
        const int arow0 = (wave * 32 + lmod) * LDA + 2 * lhalf;
        const int arow1 = arow0 + 16 * LDA;
        const int brow  = lmod * LDBT + 2 * lhalf;
#pragma unroll 4
        for (int kk = 0; kk < BK; kk += 4) {
            const v2f a0 = *(const v2f*)&As[arow0 + kk];
            const v2f a1 = *(const v2f*)&As[arow1 + kk];
            v2f b[4];
#pragma unroll
            for (int nt = 0; nt < 4; ++nt)
                b[nt] = *(const v2f*)&BsT[brow + nt * 16 * LDBT + kk];
#pragma unroll
            for (int nt = 0; nt < 4; ++nt) {
                acc[0][nt] = __builtin_amdgcn_wmma_f32_16x16x4_f32(
                    false, a0, false, b[nt], (short)0, acc[0][nt], false, false);
                acc[1][nt] = __builtin_amdgcn_wmma_f32_16x16x4_f32(
                    false, a1, false, b[nt], (short)0, acc[1][nt], false, false);
            }
        }
    }

    // Epilogue: C/D layout -> lane L holds (M = v + 8*(L>=16), N = L&15)
#pragma unroll
    for (int mt = 0; mt < 2; ++mt) {
#pragma unroll
        for (int nt = 0; nt < 4; ++nt) {
            const int col = nbase + nt * 16 + lmod;
            if (col < NDIM) {
                const int rbase = mbase + wave * 32 + mt * 16 + 8 * lhalf;
#pragma unroll
                for (int v = 0; v < 8; ++v)
                    out[(long)(rbase + v) * NDIM + col] = acc[mt][nt][v];
            }
        }
    }
}

extern "C" void kernel_launch(void* const* d_in, const int* in_sizes, int n_in,
                              void* d_out, int out_size, void* d_ws, size_t ws_size,
                              hipStream_t stream) {
    const int*   ids  = (const int*)  d_in[0];
    const float* emb  = (const float*)d_in[1];
    const float* syn  = (const float*)d_in[2];
    const float* Wsyn = (const float*)d_in[3];
    const float* bsyn = (const float*)d_in[4];
    const float* rev  = (const float*)d_in[5];
    float* outp = (float*)d_out;
    float* X    = (float*)d_ws;   // 1024*768*4 = 3 MB scratch

    build_x_kernel<<<T, 256, 0, stream>>>(ids, emb, syn, Wsyn, bsyn, X);

    dim3 grid(T / BM, (NDIM + BN - 1) / BN);   // (4, 1485); M-group fastest
    gemm_kernel<<<grid, 256, 0, stream>>>(X, rev, outp);
}